// ConvLocalShifts_37125697306853
// MI455X (gfx1250) — compile-verified
//
#include <hip/hip_runtime.h>

// ---------------------------------------------------------------------------
// ConvLocalShifts: img [B=4,T=2,C=3,H=256,W=256] f32 ->
//   shifted [B,T,C,H,W,K=49]  (77,070,336 f32)
//   mask    [B,T,1,H,W,K]     (25,690,112 f32), concatenated in d_out.
// shifted[nc,h,w,q] = x[nc, h+i-3, w+j-3] with flat=48-q, i=flat/7, j=flat%7,
// zero outside; mask = in-bounds indicator (independent of c).
// Pure bandwidth op (~411MB stores, 0 FLOPs): async global->LDS staging,
// coalesced NT b128 stores, minimized per-element index ALU.
// ---------------------------------------------------------------------------

typedef float v4f __attribute__((ext_vector_type(4)));

#define AS1 __attribute__((address_space(1)))
#define AS3 __attribute__((address_space(3)))

#if defined(__gfx1250__) && __has_builtin(__builtin_amdgcn_global_load_async_to_lds_b32)
#define USE_ASYNC_LDS 1
#else
#define USE_ASYNC_LDS 0
#endif

namespace {
constexpr int R      = 3;
constexpr int KS     = 7;            // kernel side
constexpr int K      = 49;           // KS*KS
constexpr int H      = 256;
constexpr int W      = 256;
constexpr int C      = 3;
constexpr int NC     = 24;           // B*T*C
constexpr int PITCH  = W + 2 * R + 2;     // 264 (keeps row starts 16B aligned)
constexpr int WK     = W * K;             // 12544 = 256*49
constexpr int WK4    = WK / 4;            // 3136 float4 per (nc,h) row
constexpr long SHIFT_ELEMS = (long)NC * H * WK;   // 77,070,336
constexpr int NTHREADS = 256;
}

__global__ __launch_bounds__(NTHREADS)
void ConvLocalShifts_37125697306853_kernel(const float* __restrict__ img,
                                           float* __restrict__ out)
{
    __shared__ float tile[KS * PITCH];   // 7 x 264 f32 = 7392 B

    const int h   = blockIdx.x;          // row
    const int nc  = blockIdx.y;          // fused (b,t,c)
    const int tid = threadIdx.x;

    const float* rowBase = img + (size_t)nc * H * W;

    // Prefetch the center input row (CDNA5 global_prefetch_b8 path).
    __builtin_prefetch(rowBase + (size_t)h * W + tid, 0, 3);

    // --- 1) zero the tile (pad value == 0.0f) -----------------------------
    for (int idx = tid; idx < KS * PITCH; idx += NTHREADS)
        tile[idx] = 0.0f;
    __syncthreads();   // zeros complete before async writes land on top

    // --- 2) stage 7 input rows into the tile interior (ASYNCcnt path) -----
    unsigned rowValidBits = 0u;
    #pragma unroll
    for (int i = 0; i < KS; ++i) {
        const int g = h + i - R;                  // uniform per block
        if (0 <= g && g < H) {
            rowValidBits |= (1u << i);
            const float* gp = rowBase + (size_t)g * W + tid;   // lane-coalesced
            float*       lp = &tile[i * PITCH + R + tid];
#if USE_ASYNC_LDS
            __builtin_amdgcn_global_load_async_to_lds_b32(
                (AS1 int*)gp, (AS3 int*)lp, 0, 0);
#else
            *lp = *gp;
#endif
        }
    }
#if USE_ASYNC_LDS
#if __has_builtin(__builtin_amdgcn_s_wait_asynccnt)
    __builtin_amdgcn_s_wait_asynccnt(0);
#else
    asm volatile("s_wait_asynccnt 0" ::: "memory");
#endif
#endif
    __syncthreads();

    // --- 3) stream the 49*W outputs for this (nc,h) row --------------------
    v4f* __restrict__ outS = ((v4f*)out) + (size_t)(nc * H + h) * WK4;

    if ((nc % C) == 0) {
        // channel 0 also emits the (channel-independent) mask row
        const int bt = nc / C;
        v4f* __restrict__ outM =
            ((v4f*)(out + SHIFT_ELEMS)) + (size_t)(bt * H + h) * WK4;

        for (int v = tid; v < WK4; v += NTHREADS) {  // coalesced 16B lanes
            const int f0 = v * 4;
            int w = f0 / K;                  // ONE const-div per float4
            int q = f0 - w * K;
            v4f s, m;
            #pragma unroll
            for (int e = 0; e < 4; ++e) {
                const int flat = (K - 1) - q;        // invert_order=True
                const int i    = (flat * 37) >> 8;   // == flat/7 on [0,48]
                const int j    = flat - i * KS;
                s[e] = tile[i * PITCH + w + j];      // = x[h+i-3, w+j-3] or 0
                const int col  = w + j - R;
                const bool ok  = ((rowValidBits >> i) & 1u) &&
                                 ((unsigned)col < (unsigned)W);
                m[e] = ok ? 1.0f : 0.0f;
                if (++q == K) { q = 0; ++w; }        // incremental wrap
            }
            __builtin_nontemporal_store(s, outS + v);  // write-once: bypass L2
            __builtin_nontemporal_store(m, outM + v);
        }
    } else {
        for (int v = tid; v < WK4; v += NTHREADS) {
            const int f0 = v * 4;
            int w = f0 / K;
            int q = f0 - w * K;
            v4f s;
            #pragma unroll
            for (int e = 0; e < 4; ++e) {
                const int flat = (K - 1) - q;
                const int i    = (flat * 37) >> 8;
                const int j    = flat - i * KS;
                s[e] = tile[i * PITCH + w + j];
                if (++q == K) { q = 0; ++w; }
            }
            __builtin_nontemporal_store(s, outS + v);
        }
    }
}

extern "C" void kernel_launch(void* const* d_in, const int* in_sizes, int n_in,
                              void* d_out, int out_size, void* d_ws, size_t ws_size,
                              hipStream_t stream)
{
    const float* img = (const float*)d_in[0];
    float*       out = (float*)d_out;

    dim3 grid(H, NC);        // 256 x 24 = 6144 blocks
    dim3 block(NTHREADS);    // 8 waves (wave32) per block
    hipLaunchKernelGGL(ConvLocalShifts_37125697306853_kernel,
                       grid, block, 0, stream, img, out);
}